// LinearTextEmbedding_19499151524349
// MI455X (gfx1250) — compile-verified
//
#include <hip/hip_runtime.h>
#include <hip/hip_bf16.h>

// Problem constants (from reference)
#define N_BATCH   32
#define N_BITS    1024
#define CHANNELS  16
#define WIDTH     512
#define HEIGHT    512
#define MAP       (WIDTH * HEIGHT)        // 262144, and flat_idx == i (identity scatter)

// Tiling
#define THREADS        256
#define POS_PER_BLOCK  8192               // positions of the flat map per block
#define CHUNKS         (MAP / POS_PER_BLOCK)   // 32
#define J_ITERS        (POS_PER_BLOCK / N_BITS) // 8 pattern repeats per chunk

#ifndef __has_builtin
#define __has_builtin(x) 0
#endif

// Async LDS->global store path is only meaningful (and only parseable) in the
// device pass; the host pass of the HIP TU falls back to plain stores.
#if defined(__AMDGCN__) && __has_builtin(__builtin_amdgcn_global_store_async_from_lds_b128)
#define USE_ASYNC_STORE 1
#else
#define USE_ASYNC_STORE 0
#endif

#if USE_ASYNC_STORE
typedef int v4i __attribute__((ext_vector_type(4)));
typedef __attribute__((address_space(1))) v4i gv4i;   // global-memory v4i
typedef __attribute__((address_space(3))) v4i lv4i;   // LDS v4i
#endif

__global__ __launch_bounds__(THREADS) void
LinearTextEmbedding_fill_kernel(const float* __restrict__ bits,
                                float* __restrict__ out)
{
    __shared__ float pat[N_BITS];   // thresholded pattern for this batch (4 KB)

    const int bid   = blockIdx.x;
    const int n     = bid >> 5;       // batch index      (bid / CHUNKS)
    const int chunk = bid & 31;       // chunk-in-map     (bid % CHUNKS)
    const int t     = threadIdx.x;

    // --- Stage: threshold 4 bit-values per thread into LDS (covers all 1024) ---
    const float4 b4 = reinterpret_cast<const float4*>(bits + (size_t)n * N_BITS)[t];
    float4 v;
    v.x = (fabsf(b4.x) > 0.5f) ? 1.0f : 0.0f;
    v.y = (fabsf(b4.y) > 0.5f) ? 1.0f : 0.0f;
    v.z = (fabsf(b4.z) > 0.5f) ? 1.0f : 0.0f;
    v.w = (fabsf(b4.w) > 0.5f) ? 1.0f : 0.0f;
    reinterpret_cast<float4*>(pat)[t] = v;
    __syncthreads();

    // Positions handled by this thread: pos = p0 + t*4 + j*1024.
    // Since POS_PER_BLOCK is a multiple of N_BITS, pos % N_BITS == t*4 always:
    // this thread's LDS source address is constant across all 128 stores.
    const int    p0   = chunk * POS_PER_BLOCK;
    float* const gp00 = out + ((size_t)n * CHANNELS) * MAP + p0 + t * 4;

#if USE_ASYNC_STORE
    lv4i* const lsrc = (lv4i*)(pat + t * 4);
    #pragma unroll
    for (int c = 0; c < CHANNELS; ++c) {
        float* const gpc = gp00 + (size_t)c * MAP;
        #pragma unroll
        for (int j = 0; j < J_ITERS; ++j) {
            // 16 B per lane, LDS -> memory, no VGPR data movement (ASYNCcnt).
            __builtin_amdgcn_global_store_async_from_lds_b128(
                (gv4i*)(gpc + j * N_BITS), lsrc,
                /*imm offset=*/0, /*cpol=*/0);
        }
    }
    // Drain outstanding async stores before the wave retires.
  #if __has_builtin(__builtin_amdgcn_s_wait_asynccnt)
    __builtin_amdgcn_s_wait_asynccnt(0);
  #else
    asm volatile("s_wait_asynccnt 0" ::: "memory");
  #endif
#else
    // Fallback: the thread's pattern slice is exactly its own v — plain b128 stores.
    #pragma unroll
    for (int c = 0; c < CHANNELS; ++c) {
        float* const gpc = gp00 + (size_t)c * MAP;
        #pragma unroll
        for (int j = 0; j < J_ITERS; ++j) {
            *reinterpret_cast<float4*>(gpc + j * N_BITS) = v;
        }
    }
#endif
}

extern "C" void kernel_launch(void* const* d_in, const int* in_sizes, int n_in,
                              void* d_out, int out_size, void* d_ws, size_t ws_size,
                              hipStream_t stream)
{
    (void)in_sizes; (void)n_in; (void)out_size; (void)d_ws; (void)ws_size;
    const float* bits = (const float*)d_in[0];
    float*       out  = (float*)d_out;

    const dim3 grid(N_BATCH * CHUNKS);   // 1024 blocks
    const dim3 block(THREADS);           // 256 threads = 8 wave32
    LinearTextEmbedding_fill_kernel<<<grid, block, 0, stream>>>(bits, out);
}